// GNNRecommender_85770496901303
// MI455X (gfx1250) — compile-verified
//
#include <hip/hip_runtime.h>

#define H  128
#define NU 100000
#define NP 200000
#define EV 500000
#define EL 150000

typedef float v2f __attribute__((ext_vector_type(2)));
typedef float v8f __attribute__((ext_vector_type(8)));

// ---------------------------------------------------------------------------
// Pre-sum the two lin_r weight matrices / biases that act on the same x_dst:
//   post update uses relations (0,2), user update uses relations (1,3).
// wrs: [4][H][H]  (l0-post, l0-user, l1-post, l1-user);  bs: [4][H]
// ---------------------------------------------------------------------------
__global__ void prep_weights(const float* __restrict__ W_r,
                             const float* __restrict__ b_l,
                             float* __restrict__ wrs, float* __restrict__ bs) {
  int idx = blockIdx.x * blockDim.x + threadIdx.x;
  if (idx >= 4 * H * H) return;
  int i   = idx / (H * H);          // 0..3 : [layer][ntype]
  int rem = idx % (H * H);
  int l = i >> 1, nt = i & 1;
  int ra = nt ? 1 : 0, rb = nt ? 3 : 2;
  wrs[idx] = W_r[(l * 4 + ra) * H * H + rem] + W_r[(l * 4 + rb) * H * H + rem];
  if (rem < H)
    bs[i * H + rem] = b_l[(l * 4 + ra) * H + rem] + b_l[(l * 4 + rb) * H + rem];
}

// Degree histogram (same for both layers).
__global__ void count_kernel(const int* __restrict__ dst,
                             float* __restrict__ cnt, int E) {
  int t = blockIdx.x * blockDim.x + threadIdx.x;
  if (t < E) unsafeAtomicAdd(&cnt[dst[t]], 1.0f);
}

// Edge gather + scatter-add: 32 lanes per edge, float4 per lane.
// Sum buffers fit in the 192MB L2, so atomic-adds resolve in L2.
__global__ void scatter_kernel(const float* __restrict__ x,
                               const int* __restrict__ src,
                               const int* __restrict__ dst,
                               float* __restrict__ sum, int E) {
  int t = blockIdx.x * blockDim.x + threadIdx.x;
  int e = t >> 5, lane = t & 31;
  if (e >= E) return;
  int s = src[e], d = dst[e];
  const float4 v = *(const float4*)(x + (size_t)s * H + lane * 4);
  float* p = sum + (size_t)d * H + lane * 4;
  unsafeAtomicAdd(p + 0, v.x);
  unsafeAtomicAdd(p + 1, v.y);
  unsafeAtomicAdd(p + 2, v.z);
  unsafeAtomicAdd(p + 3, v.w);
}

// ---------------------------------------------------------------------------
// Fused SAGE GEMM with LDS-staged weights.
//   out[N,128] = (sumV/cntV)@WlV^T + (sumL/cntL)@WlL^T + X@WrS^T + bias [+ReLU]
// Block = 256 threads = 8 waves, one 16-row strip per wave.
// K is processed in 4 chunks of 32; per chunk the block stages a
// 3x128x32 weight slab into LDS (float4 copies), then each wave issues
// 3 x 8 x 8 V_WMMA_F32_16X16X4_F32 with B operands from LDS.
// Row pitch 36 floats: 16B-aligned float4 stores + conflict-free float2 reads.
// ---------------------------------------------------------------------------
#define WP 36   // padded LDS row pitch (floats)

__global__ __launch_bounds__(256) void fused_sage_gemm(
    const float* __restrict__ sumV, const float* __restrict__ cntV,
    const float* __restrict__ sumL, const float* __restrict__ cntL,
    const float* __restrict__ X,
    const float* __restrict__ WlV, const float* __restrict__ WlL,
    const float* __restrict__ WrS, const float* __restrict__ bias,
    float* __restrict__ out, int nRows, int doRelu) {
  __shared__ float wlds[3][H][WP];          // 3*128*36*4 = 54 KB

  const int lane  = threadIdx.x & 31;
  const int wave  = threadIdx.x >> 5;
  const int strip = blockIdx.x * 8 + wave;
  const bool active = (strip * 16) < nRows;
  const int rowBase = active ? strip * 16 : 0;   // clamp: loads stay in-bounds
  const int m  = lane & 15;
  const int kh = (lane >> 4) << 1;               // 0 or 2: K-pair per lane half

  const float* Ag[3] = {sumV, sumL, X};
  const float* Wg[3] = {WlV, WlL, WrS};
  const float sc[3] = {1.0f / fmaxf(cntV[rowBase + m], 1.0f),
                       1.0f / fmaxf(cntL[rowBase + m], 1.0f),
                       1.0f};

  // cooperative-copy coordinates: 2 threads per weight row, 16 floats each
  const int cn = threadIdx.x >> 1;
  const int ck = (threadIdx.x & 1) * 16;

  v8f acc[8] = {};

#pragma unroll
  for (int c = 0; c < 4; ++c) {                  // K chunks of 32
    __syncthreads();                             // previous chunk fully consumed
#pragma unroll
    for (int s = 0; s < 3; ++s) {
      const float4* gs = (const float4*)(Wg[s] + (size_t)cn * H + c * 32 + ck);
      float4* ld = (float4*)(&wlds[s][cn][ck]);
      ld[0] = gs[0]; ld[1] = gs[1]; ld[2] = gs[2]; ld[3] = gs[3];
    }
    __syncthreads();                             // slab visible to all waves

#pragma unroll
    for (int s = 0; s < 3; ++s) {
      const float* arow = Ag[s] + (size_t)(rowBase + m) * H + c * 32 + kh;
      const float scl = sc[s];
#pragma unroll
      for (int kk = 0; kk < 32; kk += 4) {
        v2f a = *(const v2f*)(arow + kk);
        a = a * scl;                             // fold mean-normalization
#pragma unroll
        for (int nt = 0; nt < 8; ++nt) {
          v2f b = *(const v2f*)(&wlds[s][nt * 16 + m][kk + kh]);
          acc[nt] = __builtin_amdgcn_wmma_f32_16x16x4_f32(
              false, a, false, b, (short)0, acc[nt], false, false);
        }
      }
    }
  }

  if (!active) return;                           // wave-uniform exit
  const int rhalf = (lane >> 4) << 3;            // C/D: VGPR r -> row r+8*(lane>>4)
#pragma unroll
  for (int nt = 0; nt < 8; ++nt) {
    const int col = nt * 16 + m;
    const float bv = bias[col];
#pragma unroll
    for (int r = 0; r < 8; ++r) {
      float v = acc[nt][r] + bv;
      if (doRelu) v = fmaxf(v, 0.0f);
      out[(size_t)(rowBase + r + rhalf) * H + col] = v;
    }
  }
}

extern "C" void kernel_launch(void* const* d_in, const int* in_sizes, int n_in,
                              void* d_out, int out_size, void* d_ws, size_t ws_size,
                              hipStream_t stream) {
  const float* emb_user = (const float*)d_in[0];
  const float* emb_post = (const float*)d_in[1];
  const float* W_l      = (const float*)d_in[2];   // [2,4,H,H]
  const float* b_l      = (const float*)d_in[3];   // [2,4,H]
  const float* W_r      = (const float*)d_in[4];   // [2,4,H,H]
  const int* ev_s = (const int*)d_in[5];
  const int* ev_d = (const int*)d_in[6];
  const int* el_s = (const int*)d_in[7];
  const int* el_d = (const int*)d_in[8];

  float* ws = (float*)d_ws;
  size_t off = 0;
  float* sum_vp = ws + off; off += (size_t)NP * H;
  float* sum_lp = ws + off; off += (size_t)NP * H;
  float* sum_vu = ws + off; off += (size_t)NU * H;
  float* sum_lu = ws + off; off += (size_t)NU * H;
  size_t sums_elems = off;
  float* cnt_vp = ws + off; off += NP;
  float* cnt_lp = ws + off; off += NP;
  float* cnt_vu = ws + off; off += NU;
  float* cnt_lu = ws + off; off += NU;
  size_t sums_cnts_elems = off;
  float* h_user = ws + off; off += (size_t)NU * H;
  float* h_post = ws + off; off += (size_t)NP * H;
  float* wrs    = ws + off; off += 4 * H * H;
  float* bs     = ws + off; off += 4 * H;

  float* out_xu = (float*)d_out;
  float* out_xp = (float*)d_out + (size_t)NU * H;

  // Weight pre-sums (tiny).
  prep_weights<<<(4 * H * H + 255) / 256, 256, 0, stream>>>(W_r, b_l, wrs, bs);
  // Zero sums + counts.
  hipMemsetAsync(ws, 0, sums_cnts_elems * sizeof(float), stream);
  // Degree counts (shared by both layers).
  count_kernel<<<(EV + 255) / 256, 256, 0, stream>>>(ev_d, cnt_vp, EV);
  count_kernel<<<(EL + 255) / 256, 256, 0, stream>>>(el_d, cnt_lp, EL);
  count_kernel<<<(EV + 255) / 256, 256, 0, stream>>>(ev_s, cnt_vu, EV);
  count_kernel<<<(EL + 255) / 256, 256, 0, stream>>>(el_s, cnt_lu, EL);

  const int sc_ev = (EV * 32 + 255) / 256;
  const int sc_el = (EL * 32 + 255) / 256;

  // ---- Layer 1 edge phase ----
  scatter_kernel<<<sc_ev, 256, 0, stream>>>(emb_user, ev_s, ev_d, sum_vp, EV);
  scatter_kernel<<<sc_el, 256, 0, stream>>>(emb_user, el_s, el_d, sum_lp, EL);
  scatter_kernel<<<sc_ev, 256, 0, stream>>>(emb_post, ev_d, ev_s, sum_vu, EV);
  scatter_kernel<<<sc_el, 256, 0, stream>>>(emb_post, el_d, el_s, sum_lu, EL);

  // ---- Layer 1 fused GEMMs (+ReLU) ----
  fused_sage_gemm<<<(NP / 16 + 7) / 8, 256, 0, stream>>>(
      sum_vp, cnt_vp, sum_lp, cnt_lp, emb_post,
      W_l + 0 * H * H, W_l + 2 * H * H, wrs + 0 * H * H, bs + 0 * H,
      h_post, NP, 1);
  fused_sage_gemm<<<(NU / 16 + 7) / 8, 256, 0, stream>>>(
      sum_vu, cnt_vu, sum_lu, cnt_lu, emb_user,
      W_l + 1 * H * H, W_l + 3 * H * H, wrs + 1 * H * H, bs + 1 * H,
      h_user, NU, 1);

  // ---- Layer 2 edge phase ----
  hipMemsetAsync(ws, 0, sums_elems * sizeof(float), stream);
  scatter_kernel<<<sc_ev, 256, 0, stream>>>(h_user, ev_s, ev_d, sum_vp, EV);
  scatter_kernel<<<sc_el, 256, 0, stream>>>(h_user, el_s, el_d, sum_lp, EL);
  scatter_kernel<<<sc_ev, 256, 0, stream>>>(h_post, ev_d, ev_s, sum_vu, EV);
  scatter_kernel<<<sc_el, 256, 0, stream>>>(h_post, el_d, el_s, sum_lu, EL);

  // ---- Layer 2 fused GEMMs -> output (no ReLU) ----
  fused_sage_gemm<<<(NP / 16 + 7) / 8, 256, 0, stream>>>(
      sum_vp, cnt_vp, sum_lp, cnt_lp, h_post,
      W_l + (4 + 0) * H * H, W_l + (4 + 2) * H * H, wrs + 2 * H * H, bs + 2 * H,
      out_xp, NP, 0);
  fused_sage_gemm<<<(NU / 16 + 7) / 8, 256, 0, stream>>>(
      sum_vu, cnt_vu, sum_lu, cnt_lu, h_user,
      W_l + (4 + 1) * H * H, W_l + (4 + 3) * H * H, wrs + 3 * H * H, bs + 3 * H,
      out_xu, NU, 0);

  (void)in_sizes; (void)n_in; (void)out_size; (void)ws_size;
}